// ProteinFoldingPredictor_51539608079
// MI455X (gfx1250) — compile-verified
//
#include <hip/hip_runtime.h>

// Problem constants
#define SEQ  20
#define HID  512
#define NROW 8192            // B*L = 4*2048
#define LLEN 2048
#define ROWS_PER_WG 32       // 2 row-tiles of 16; ~140KB LDS -> 2 WGs per WGP
#define H1S  516             // padded LDS row stride (bank-step 4 -> conflict free)

typedef __attribute__((ext_vector_type(2))) float v2f;
typedef __attribute__((ext_vector_type(4))) float f4v;
typedef __attribute__((ext_vector_type(8))) float v8f;

// ---------------------------------------------------------------------------
// Fused encoder: h1 = relu(x@W1+b1); h2 = relu(h1@W2+b2); heads from h2.
// One workgroup = 256 threads = 8 waves, handles 32 rows.
// Wave w owns col-tiles 4w..4w+3 (16 cols each) with 2 row-tile accumulators.
// ---------------------------------------------------------------------------
__global__ __launch_bounds__(256) void pf_encoder_kernel(
    const float* __restrict__ x,   const float* __restrict__ W1, const float* __restrict__ b1,
    const float* __restrict__ W2,  const float* __restrict__ b2,
    const float* __restrict__ Wss, const float* __restrict__ bss,
    const float* __restrict__ Wang,const float* __restrict__ bang,
    const float* __restrict__ Wc,
    float* __restrict__ out_ss, float* __restrict__ out_ang,
    float* __restrict__ ci_g, float* __restrict__ cj_g)
{
    extern __shared__ float smem[];
    float* lds_x  = smem;                        // 32*20
    float* lds_h1 = lds_x  + ROWS_PER_WG * SEQ;  // 32*516
    float* lds_h2 = lds_h1 + ROWS_PER_WG * H1S;  // 32*516
    float* lds_hd = lds_h2 + ROWS_PER_WG * H1S;  // 32*8*8 head partials

    const int tid   = threadIdx.x;
    const int lane  = tid & 31;
    const int wv    = tid >> 5;      // wave 0..7
    const int mlane = lane & 15;     // M index (A) / N index (B,C)
    const int hi    = lane >> 4;     // half-wave select
    const int row0  = blockIdx.x * ROWS_PER_WG;

    // ---- stage x tile into LDS via gfx1250 async global->LDS copy ----
    // x tile is contiguous in memory: 32*20 = 640 dwords starting at row0*SEQ.
    // GVS mode: SGPR 64-bit base + per-lane u32 byte offset; VDST = LDS byte addr.
    {
        const float* gsrc = x + (size_t)row0 * SEQ;
        for (int idx = tid; idx < ROWS_PER_WG * SEQ; idx += 256) {
            unsigned ldsb = (unsigned)(size_t)(void*)(lds_x + idx);  // LDS byte address
            unsigned goff = (unsigned)idx * 4u;                      // global byte offset
            asm volatile("global_load_async_to_lds_b32 %0, %1, %2"
                         :: "v"(ldsb), "v"(goff), "s"(gsrc) : "memory");
        }
        asm volatile("s_wait_asynccnt 0" ::: "memory");
    }
    __syncthreads();

    // ---- GEMM1: h1[32x512] = relu(x @ W1 + b1), K = 20 (5 steps of 4) ----
    for (int ct = 0; ct < 4; ++ct) {
        const int ncol = (wv * 4 + ct) * 16 + mlane;
        v8f acc0 = {}; v8f acc1 = {};
#pragma unroll
        for (int kb = 0; kb < SEQ; kb += 4) {
            const int k0 = kb + 2 * hi;
            v2f bv; bv.x = W1[(size_t)k0 * HID + ncol];
                    bv.y = W1[(size_t)(k0 + 1) * HID + ncol];
            v2f a0; a0.x = lds_x[mlane * SEQ + k0];
                    a0.y = lds_x[mlane * SEQ + k0 + 1];
            v2f a1; a1.x = lds_x[(16 + mlane) * SEQ + k0];
                    a1.y = lds_x[(16 + mlane) * SEQ + k0 + 1];
            acc0 = __builtin_amdgcn_wmma_f32_16x16x4_f32(false, a0, false, bv, (short)0, acc0, false, false);
            acc1 = __builtin_amdgcn_wmma_f32_16x16x4_f32(false, a1, false, bv, (short)0, acc1, false, false);
        }
        const float bias = b1[ncol];
#pragma unroll
        for (int r = 0; r < 8; ++r) {
            float v0 = acc0[r] + bias; v0 = v0 > 0.f ? v0 : 0.f;
            float v1 = acc1[r] + bias; v1 = v1 > 0.f ? v1 : 0.f;
            lds_h1[(8 * hi + r) * H1S + ncol]        = v0;
            lds_h1[(16 + 8 * hi + r) * H1S + ncol]   = v1;
        }
    }
    __syncthreads();

    // ---- GEMM2: h2[32x512] = relu(h1 @ W2 + b2), K = 512 (128 steps) ----
    for (int ct = 0; ct < 4; ++ct) {
        const int ncol = (wv * 4 + ct) * 16 + mlane;
        const float* __restrict__ w2p = W2 + ncol;
        v8f acc0 = {}; v8f acc1 = {};
#pragma unroll 4
        for (int kb = 0; kb < HID; kb += 4) {
            const int k0 = kb + 2 * hi;
            if (kb + 16 < HID)
                __builtin_prefetch(w2p + (size_t)(kb + 16) * HID, 0, 3);
            v2f bv; bv.x = w2p[(size_t)k0 * HID];
                    bv.y = w2p[(size_t)(k0 + 1) * HID];
            v2f a0; a0.x = lds_h1[mlane * H1S + k0];
                    a0.y = lds_h1[mlane * H1S + k0 + 1];
            v2f a1; a1.x = lds_h1[(16 + mlane) * H1S + k0];
                    a1.y = lds_h1[(16 + mlane) * H1S + k0 + 1];
            acc0 = __builtin_amdgcn_wmma_f32_16x16x4_f32(false, a0, false, bv, (short)0, acc0, false, false);
            acc1 = __builtin_amdgcn_wmma_f32_16x16x4_f32(false, a1, false, bv, (short)0, acc1, false, false);
        }
        const float bias = b2[ncol];
#pragma unroll
        for (int r = 0; r < 8; ++r) {
            float v0 = acc0[r] + bias; v0 = v0 > 0.f ? v0 : 0.f;
            float v1 = acc1[r] + bias; v1 = v1 > 0.f ? v1 : 0.f;
            lds_h2[(8 * hi + r) * H1S + ncol]        = v0;
            lds_h2[(16 + 8 * hi + r) * H1S + ncol]   = v1;
        }
    }
    __syncthreads();

    // ---- Heads: 8 threads per row, each reduces 64 of 512 k-values ----
    {
        const int row  = tid >> 3;   // 0..31
        const int part = tid & 7;
        const float* __restrict__ h = lds_h2 + row * H1S;
        float s0 = 0.f, s1 = 0.f, s2 = 0.f, a0 = 0.f, a1 = 0.f, ci = 0.f, cj = 0.f;
        const int kbeg = part * 64;
#pragma unroll 8
        for (int k = kbeg; k < kbeg + 64; ++k) {
            const float hv = h[k];
            s0 += hv * Wss[k * 3 + 0];
            s1 += hv * Wss[k * 3 + 1];
            s2 += hv * Wss[k * 3 + 2];
            a0 += hv * Wang[k * 2 + 0];
            a1 += hv * Wang[k * 2 + 1];
            ci += hv * Wc[k];
            cj += hv * Wc[HID + k];
        }
        float* sc = lds_hd + (row * 8 + part) * 8;
        sc[0] = s0; sc[1] = s1; sc[2] = s2; sc[3] = a0; sc[4] = a1; sc[5] = ci; sc[6] = cj;
    }
    __syncthreads();

    if (tid < ROWS_PER_WG) {
        float acc[7] = {0.f, 0.f, 0.f, 0.f, 0.f, 0.f, 0.f};
        for (int p = 0; p < 8; ++p) {
            const float* sc = lds_hd + (tid * 8 + p) * 8;
#pragma unroll
            for (int c = 0; c < 7; ++c) acc[c] += sc[c];
        }
        const int grow = row0 + tid;
        // softmax over 3 logits
        const float l0 = acc[0] + bss[0], l1 = acc[1] + bss[1], l2 = acc[2] + bss[2];
        const float m  = fmaxf(l0, fmaxf(l1, l2));
        const float e0 = __expf(l0 - m), e1 = __expf(l1 - m), e2 = __expf(l2 - m);
        const float inv = 1.f / (e0 + e1 + e2);
        out_ss[grow * 3 + 0] = e0 * inv;
        out_ss[grow * 3 + 1] = e1 * inv;
        out_ss[grow * 3 + 2] = e2 * inv;
        out_ang[grow * 2 + 0] = acc[3] + bang[0];
        out_ang[grow * 2 + 1] = acc[4] + bang[1];
        ci_g[grow] = acc[5];
        cj_g[grow] = acc[6];
    }
}

// ---------------------------------------------------------------------------
// Contact map: out[gi][j] = sigmoid(ci[gi] + cj[batch_base + j] + bc)
// Pure streaming-store kernel: one row (2048 floats) per block, float4 stores.
// 67 MB of stores -> HBM-bandwidth bound (~3us at 23.3 TB/s).
// ---------------------------------------------------------------------------
__global__ __launch_bounds__(256) void pf_contact_kernel(
    const float* __restrict__ ci, const float* __restrict__ cj,
    const float* __restrict__ bc, float* __restrict__ out)
{
    const int gi   = blockIdx.x;            // 0..8191
    const int jb   = (gi >> 11) << 11;      // batch base: b * 2048
    const float base = ci[gi] + bc[0];
    f4v* __restrict__ dst = (f4v*)(out + (size_t)gi * LLEN);
    const f4v* __restrict__ cjv = (const f4v*)(cj + jb);
#pragma unroll
    for (int it = 0; it < 2; ++it) {
        const int t = threadIdx.x + it * 256;   // 512 float4 per row
        f4v v = cjv[t];
        f4v o;
        o.x = 1.f / (1.f + __expf(-(base + v.x)));
        o.y = 1.f / (1.f + __expf(-(base + v.y)));
        o.z = 1.f / (1.f + __expf(-(base + v.z)));
        o.w = 1.f / (1.f + __expf(-(base + v.w)));
        dst[t] = o;
    }
}

// ---------------------------------------------------------------------------
extern "C" void kernel_launch(void* const* d_in, const int* in_sizes, int n_in,
                              void* d_out, int out_size, void* d_ws, size_t ws_size,
                              hipStream_t stream)
{
    const float* x    = (const float*)d_in[0];
    const float* W1   = (const float*)d_in[1];
    const float* b1   = (const float*)d_in[2];
    const float* W2   = (const float*)d_in[3];
    const float* b2   = (const float*)d_in[4];
    const float* Wss  = (const float*)d_in[5];
    const float* bss  = (const float*)d_in[6];
    const float* Wang = (const float*)d_in[7];
    const float* bang = (const float*)d_in[8];
    const float* Wc   = (const float*)d_in[9];
    const float* bc   = (const float*)d_in[10];

    float* out      = (float*)d_out;
    float* out_ss   = out;                       // [8192,3]
    float* out_ang  = out + NROW * 3;            // [8192,2]
    float* out_con  = out + NROW * 3 + NROW * 2; // [4,2048,2048]

    float* ci = (float*)d_ws;                    // [8192]
    float* cj = ci + NROW;                       // [8192]

    const size_t shmem =
        (ROWS_PER_WG * SEQ + 2 * ROWS_PER_WG * H1S + ROWS_PER_WG * 64) * sizeof(float);

    pf_encoder_kernel<<<NROW / ROWS_PER_WG, 256, shmem, stream>>>(
        x, W1, b1, W2, b2, Wss, bss, Wang, bang, Wc,
        out_ss, out_ang, ci, cj);

    pf_contact_kernel<<<NROW, 256, 0, stream>>>(ci, cj, bc, out_con);
}